// CustomLoss_49263274885694
// MI455X (gfx1250) — compile-verified
//
#include <hip/hip_runtime.h>

#define B_DIM 512
#define S_DIM 4096
#define C_DIM 8
#define PENALTY_WEIGHT 0.1f

typedef __attribute__((ext_vector_type(2))) float v2f;
typedef __attribute__((ext_vector_type(8))) float v8f;

#if defined(__has_builtin)
#if __has_builtin(__builtin_amdgcn_global_load_async_to_lds_b128)
#define HAVE_ASYNC_LDS 1
#endif
#endif

// ---------------------------------------------------------------------------
// Wave32 sum-reduction through the matrix unit:
//   A (16x4 f32) holds one value per lane in A-VGPR0, zeros in A-VGPR1.
//   B = all ones (layout-independent), C = 0:
//   D[m][n] = sum_k A[m][k]  ->  row totals replicated across columns.
//   Summing the 8 D VGPRs gives rows 0..7 total on lanes 0..15 and rows 8..15
//   total on lanes 16..31; two shuffles complete the 32-lane reduction.
// EXEC must be all ones at the call site.
// ---------------------------------------------------------------------------
__device__ __forceinline__ float wave_reduce_sum_wmma(float partial) {
    v2f a;    a.x = partial; a.y = 0.0f;
    v2f ones; ones.x = 1.0f; ones.y = 1.0f;
    v8f c = {};
    v8f d = __builtin_amdgcn_wmma_f32_16x16x4_f32(
        /*neg_a=*/false, a, /*neg_b=*/false, ones,
        /*c_mod=*/(short)0, c, /*reuse_a=*/false, /*reuse_b=*/false);
    float s8 = d[0] + d[1] + d[2] + d[3] + d[4] + d[5] + d[6] + d[7];
    float lo = __shfl(s8, 0, 32);   // rows 0..7 total
    float hi = __shfl(s8, 16, 32);  // rows 8..15 total
    return lo + hi;
}

// ---------------------------------------------------------------------------
// Zero the workspace accumulators (graph-replayed every call).
// ws layout (dwords): [0]=ce_sum(f32), [1]=nonzero_count(u32), [2]=pen_total(i32)
// ---------------------------------------------------------------------------
__global__ void init_ws_kernel(unsigned int* ws) {
    if (threadIdx.x < 4) ws[threadIdx.x] = 0u;
}

// ---------------------------------------------------------------------------
// Weighted cross-entropy partial sums + nonzero-target count.
// One thread per (b,s): two b128 loads of the 8 logits (32B contiguous per
// lane -> 1KB per wave, perfectly coalesced).
// ---------------------------------------------------------------------------
__global__ __launch_bounds__(256) void ce_kernel(
    const float* __restrict__ logits,
    const int*   __restrict__ targets,
    const float* __restrict__ w,
    float* __restrict__ ce_sum,
    unsigned int* __restrict__ nz_count) {

    const int idx  = blockIdx.x * blockDim.x + threadIdx.x;   // < B*S exactly
    const int lane = threadIdx.x & 31;
    const int wid  = threadIdx.x >> 5;

    const float4* lg = (const float4*)logits + (size_t)idx * 2;
    float4 x0 = lg[0];
    float4 x1 = lg[1];
    int t = targets[idx];

    float m = fmaxf(fmaxf(fmaxf(x0.x, x0.y), fmaxf(x0.z, x0.w)),
                    fmaxf(fmaxf(x1.x, x1.y), fmaxf(x1.z, x1.w)));
    float se = __expf(x0.x - m) + __expf(x0.y - m) +
               __expf(x0.z - m) + __expf(x0.w - m) +
               __expf(x1.x - m) + __expf(x1.y - m) +
               __expf(x1.z - m) + __expf(x1.w - m);
    float lse = __logf(se);

    // branchless gather of logits[t] (t in 0..7) via cndmask chain
    float xt = (t < 4)
        ? ((t < 2) ? ((t == 0) ? x0.x : x0.y) : ((t == 2) ? x0.z : x0.w))
        : ((t < 6) ? ((t == 4) ? x1.x : x1.y) : ((t == 6) ? x1.z : x1.w));

    float contrib = (xt - m - lse) * w[t];

    // wave reduction through the matrix unit (all lanes active here)
    float wsum = wave_reduce_sum_wmma(contrib);
    unsigned long long bal = __ballot(t != 0);
    int wcnt = __popcll(bal);

    __shared__ float        sdata[8];
    __shared__ unsigned int scnt[8];
    if (lane == 0) { sdata[wid] = wsum; scnt[wid] = (unsigned int)wcnt; }
    __syncthreads();
    if (threadIdx.x == 0) {
        float bs = 0.0f; unsigned int bc = 0u;
        #pragma unroll
        for (int i = 0; i < 8; ++i) { bs += sdata[i]; bc += scnt[i]; }
        atomicAdd(ce_sum, bs);
        atomicAdd(nz_count, bc);
    }
}

// ---------------------------------------------------------------------------
// Structural penalty: one wave per batch row.
// Closed form of the clamped bracket counter:
//   pen_unmatched + final_counter = S_N - 2*min(0, min_j S_j)
// where S_j are prefix sums of delta = (s==1) - (s==2).
// Pair bonuses (2/3/4 with clamped lookahead indices) are purely local.
// Row staged in LDS via the gfx1250 async global->LDS path (ASYNCcnt),
// skipping the VGPR round-trip; falls back to a b128 copy loop if the
// builtin is unavailable.
// ---------------------------------------------------------------------------
__global__ __launch_bounds__(32) void pen_kernel(
    const int* __restrict__ structures,   // [B, S]
    int* __restrict__ pen_total) {

    __shared__ __align__(16) int row[S_DIM];
    const int b    = blockIdx.x;
    const int lane = threadIdx.x;

    // stage row: 4096 ints = 1024 int4, 32 per lane
    const int4* src = (const int4*)(structures + (size_t)b * S_DIM);
    int4* dst = (int4*)row;
#if HAVE_ASYNC_LDS
    // builtin prototype: void (*)(v4i AS1* src, v4i AS3* dst, imm off, imm cpol)
    typedef int v4i_t __attribute__((vector_size(16)));
    typedef __attribute__((address_space(1))) v4i_t* gasync_p;
    typedef __attribute__((address_space(3))) v4i_t* lasync_p;
    #pragma unroll
    for (int k = 0; k < S_DIM / 4 / 32; ++k) {            // 32 async b128 ops
        const int e = k * 32 + lane;                      // int4 index
        __builtin_amdgcn_global_load_async_to_lds_b128(
            (gasync_p)(void*)(src + e), (lasync_p)(dst + e),
            /*offset=*/0, /*cpol=*/0);
    }
#if defined(__has_builtin) && __has_builtin(__builtin_amdgcn_s_wait_asynccnt)
    __builtin_amdgcn_s_wait_asynccnt(0);
#else
    asm volatile("s_wait_asynccnt 0x0" ::: "memory");
#endif
    __syncthreads();
#else
    #pragma unroll 4
    for (int k = lane; k < S_DIM / 4; k += 32) dst[k] = src[k];
    __syncthreads();
#endif

    const int base = lane * (S_DIM / 32);   // 128 elements per lane
    int dsum = 0;            // running delta sum within chunk
    int dminp = 0x7fffffff;  // min prefix within chunk
    int pp = 0;              // local pair bonuses

    for (int j = 0; j < S_DIM / 32; ++j) {
        const int i  = base + j;
        const int si = row[i];
        const int lp = (si == 1);
        const int rp = (si == 2);
        if (lp) {
            const int rp1 = (row[min(i + 1, S_DIM - 1)] == 2);
            const int d1  = (row[min(i + 1, S_DIM - 2)] == 3);
            const int rp2 = (row[min(i + 2, S_DIM - 1)] == 2);
            const int d2  = (row[min(i + 2, S_DIM - 2)] == 3);
            const int rp3 = (row[min(i + 3, S_DIM - 1)] == 2);
            pp += 2 * rp1 + 3 * (d1 * rp2) + 4 * (d1 * d2 * rp3);
        }
        dsum += lp - rp;
        dminp = min(dminp, dsum);
    }

    // inclusive shuffle scan of per-lane delta sums
    int incl = dsum;
    #pragma unroll
    for (int sh = 1; sh < 32; sh <<= 1) {
        int v = __shfl_up(incl, sh, 32);
        if (lane >= sh) incl += v;
    }
    const int excl  = incl - dsum;          // exclusive prefix (chunk offset)
    const int total = __shfl(incl, 31, 32); // S_N for the row

    int gmin = excl + dminp;                // global prefix min candidate
    #pragma unroll
    for (int sh = 16; sh >= 1; sh >>= 1)
        gmin = min(gmin, __shfl_xor(gmin, sh, 32));

    int psum = pp;
    #pragma unroll
    for (int sh = 16; sh >= 1; sh >>= 1)
        psum += __shfl_xor(psum, sh, 32);

    if (lane == 0) {
        const int pen = psum + total - 2 * min(0, gmin);
        atomicAdd(pen_total, pen);
    }
}

// ---------------------------------------------------------------------------
// Final scalar:
//   ce_loss = -ce_sum / (B*S)
//   penalty = mean_b(pen_b) / (nz/B) = pen_total / nz
// ---------------------------------------------------------------------------
__global__ void fin_kernel(const float* __restrict__ ce_sum,
                           const unsigned int* __restrict__ nz_count,
                           const int* __restrict__ pen_total,
                           float* __restrict__ out) {
    const float ce  = -ce_sum[0] / (float)(B_DIM * S_DIM);
    const float pen = (float)pen_total[0] / (float)nz_count[0];
    out[0] = ce + PENALTY_WEIGHT * pen;
}

extern "C" void kernel_launch(void* const* d_in, const int* in_sizes, int n_in,
                              void* d_out, int out_size, void* d_ws, size_t ws_size,
                              hipStream_t stream) {
    const float* logits     = (const float*)d_in[0];
    const int*   targets    = (const int*)d_in[1];
    const int*   structures = (const int*)d_in[2];   // [B,S,1] contiguous
    const float* ce_weights = (const float*)d_in[3];
    float* out = (float*)d_out;

    unsigned int* ws  = (unsigned int*)d_ws;
    float*        ce_sum    = (float*)d_ws;                 // dword 0
    unsigned int* nz_count  = (unsigned int*)d_ws + 1;      // dword 1
    int*          pen_total = (int*)d_ws + 2;               // dword 2

    init_ws_kernel<<<1, 32, 0, stream>>>(ws);

    const int n_elems = B_DIM * S_DIM;                       // 2,097,152
    ce_kernel<<<n_elems / 256, 256, 0, stream>>>(
        logits, targets, ce_weights, ce_sum, nz_count);

    pen_kernel<<<B_DIM, 32, 0, stream>>>(structures, pen_total);

    fin_kernel<<<1, 1, 0, stream>>>(ce_sum, nz_count, pen_total, out);
}